// VADetector_9225589752323
// MI455X (gfx1250) — compile-verified
//
#include <hip/hip_runtime.h>

// Viterbi detector, 16 states, MEM=4, T=16384, B=256.
//
// MI455X design:
//  - Latency-bound serial ACS over 16388 steps; FLOPs (~0.27G) and bytes
//    (~50MB) are negligible vs 23.3 TB/s HBM. One batch per wave32 lane:
//    all 16 path metrics in VGPRs, zero cross-lane traffic.
//  - y^2 term of the branch metric is a uniform per-step offset -> dropped.
//    Metric = fma(c1[s], y, m[s>>1] + c0[s]).
//  - States 2q,2q+1 share predecessor pair (q,q+8): 8 min+decision/step,
//    decisions = sign(p[q+8]-p[q]) packed 8b/step, 4 steps per b32 store.
//  - CDNA5 async engine: y staged into LDS in 64-step tiles with
//    GLOBAL_LOAD_ASYNC_TO_LDS_B128 (ASYNCcnt), triple-buffered; consume via
//    ds_load_b128; one s_wait_asynccnt per 64 steps. INST_OFFSET is added to
//    both global and LDS addresses, so one vaddr/vdst pair serves a tile.
//  - LDS rows padded to 272B so ds_load_b128 lane addresses spread banks.

#define NSTATES 16
#define MEML    4
#define TLEN    16384
#define BATCH   256
#define TPM     (TLEN + MEML)     // 16388 time steps
#define T4      (TPM / 4)         // 4097 packed float4 groups
#define YTILE_G 16                // float4 groups per tile per lane
#define NTILES  256               // 256*16 = 4096 groups; 1 tail group direct
#define ROWB    (YTILE_G * 16 + 16)   // 272 B per lane row (16B pad)
#define BUFB    (32 * ROWB)           // 8704 B per buffer

__device__ __forceinline__ void issue_tile_async(const char* gbase,
                                                 unsigned voff,
                                                 unsigned ldsrow)
{
#pragma unroll
    for (int g = 0; g < YTILE_G; ++g) {
        // vdst = LDS byte address VGPR, vaddr = 32-bit global offset VGPR,
        // saddr = 64-bit base SGPR pair (GVS mode). offset applies to both.
        asm volatile("global_load_async_to_lds_b128 %0, %1, %2 offset:%3"
                     :: "v"(ldsrow), "v"(voff), "s"(gbase), "i"(g * 16)
                     : "memory");
    }
}

__device__ __forceinline__ void wait_async_le16()
{
    asm volatile("s_wait_asynccnt 0x10" ::: "memory");
}

__device__ __forceinline__ void wait_async_0()
{
    asm volatile("s_wait_asynccnt 0x0" ::: "memory");
}

__global__ __launch_bounds__(32)
void va_forward(const float* __restrict__ y,
                const float* __restrict__ h,
                unsigned*    __restrict__ dec)
{
    __shared__ unsigned char smem[3 * BUFB];   // 26112 B of LDS

    const int lane = threadIdx.x;
    const int b    = blockIdx.x * 32 + lane;   // batch element for this lane

    // Per-state branch-metric constants: e[s] = sum_i (1-2*bit_i(s)) * h[i]
    const float h0 = h[0], h1 = h[1], h2 = h[2], h3 = h[3];
    float c0[NSTATES], c1[NSTATES];
#pragma unroll
    for (int s = 0; s < NSTATES; ++s) {
        float e = ((s & 1) ? -h0 : h0) + ((s & 2) ? -h1 : h1)
                + ((s & 4) ? -h2 : h2) + ((s & 8) ? -h3 : h3);
        c1[s] = -2.0f * e;
        c0[s] = e * e;
    }

    float p[NSTATES];
#pragma unroll
    for (int s = 0; s < NSTATES; ++s) p[s] = 0.0f;

    unsigned* __restrict__ drow = dec + (size_t)b * T4;

    const char*    gbase = (const char*)y;
    const unsigned voff0 = (unsigned)((size_t)b * TPM * 4u); // lane row base (bytes)
    const unsigned myrow = (unsigned)(uintptr_t)&smem[0] + (unsigned)lane * ROWB;

    // Prologue: launch DMA for tiles 0 and 1.
    issue_tile_async(gbase, voff0,        myrow + 0u * BUFB);
    issue_tile_async(gbase, voff0 + 256u, myrow + 1u * BUFB);

    for (int k = 0; k < NTILES; ++k) {
        // Oldest 16 asyncs (tile k) must have landed. Async loads complete
        // in order, so cnt<=16 == tile k done while tile k+1 still flies.
        if (k == NTILES - 1) wait_async_0(); else wait_async_le16();

        // Refill the buffer consumed one tile ago (tile k-1) (its ds_loads completed:
        // their data was consumed last iteration).
        if (k + 2 < NTILES)
            issue_tile_async(gbase, voff0 + (unsigned)(k + 2) * 256u,
                             myrow + (unsigned)((k + 2) % 3) * BUFB);

        const float4* lrow =
            (const float4*)(smem + (unsigned)(k % 3) * BUFB + (unsigned)lane * ROWB);

#pragma unroll 4
        for (int g = 0; g < YTILE_G; ++g) {
            const float4 yv = lrow[g];        // ds_load_b128
            unsigned dmask = 0;
#pragma unroll
            for (int i = 0; i < 4; ++i) {
                const float yt = (i == 0) ? yv.x : (i == 1) ? yv.y
                               : (i == 2) ? yv.z : yv.w;
                float m[8];
#pragma unroll
                for (int q = 0; q < 8; ++q) {
                    const float a  = p[q];
                    const float bb = p[q + 8];
                    m[q] = fminf(a, bb);
                    // j = 1 iff p[q+8] < p[q]; ties -> 0 (argmin-first)
                    const unsigned j = __float_as_uint(bb - a) >> 31;
                    dmask |= j << (q + 8 * i);
                }
#pragma unroll
                for (int s = 0; s < NSTATES; ++s)
                    p[s] = fmaf(c1[s], yt, m[s >> 1] + c0[s]);
            }
            drow[k * YTILE_G + g] = dmask;
        }
    }

    // Tail: group t4 = 4096 (steps 16384..16387), direct global load.
    {
        const float4* yrow4 = (const float4*)(y + (size_t)b * TPM);
        const float4 yv = yrow4[T4 - 1];
        unsigned dmask = 0;
#pragma unroll
        for (int i = 0; i < 4; ++i) {
            const float yt = (i == 0) ? yv.x : (i == 1) ? yv.y
                           : (i == 2) ? yv.z : yv.w;
            float m[8];
#pragma unroll
            for (int q = 0; q < 8; ++q) {
                const float a  = p[q];
                const float bb = p[q + 8];
                m[q] = fminf(a, bb);
                const unsigned j = __float_as_uint(bb - a) >> 31;
                dmask |= j << (q + 8 * i);
            }
#pragma unroll
            for (int s = 0; s < NSTATES; ++s)
                p[s] = fmaf(c1[s], yt, m[s >> 1] + c0[s]);
        }
        drow[T4 - 1] = dmask;
    }
}

__global__ __launch_bounds__(32)
void va_traceback(const unsigned* __restrict__ dec,
                  float*          __restrict__ out)
{
    const int b = blockIdx.x * 32 + threadIdx.x;
    const unsigned* __restrict__ drow = dec + (size_t)b * T4;
    float4*         __restrict__ orow = (float4*)(out + (size_t)b * TLEN);

    unsigned state = 0;  // reference starts traceback from state 0

    // Tail group t = 16384..16387 (beyond T): update state, emit nothing.
    {
        const unsigned d4 = drow[T4 - 1];
#pragma unroll
        for (int i = 3; i >= 0; --i) {
            const unsigned q = state >> 1;
            const unsigned j = (d4 >> (8 * i + q)) & 1u;
            state = q | (j << 3);   // prev = (s>>1) + j*8 ; bit = j
        }
    }
    for (int t4 = T4 - 2; t4 >= 0; --t4) {
        const unsigned d4 = drow[t4];
        float bits[4];
#pragma unroll
        for (int i = 3; i >= 0; --i) {
            const unsigned q = state >> 1;
            const unsigned j = (d4 >> (8 * i + q)) & 1u;
            state = q | (j << 3);
            bits[i] = (float)j;
        }
        orow[t4] = make_float4(bits[0], bits[1], bits[2], bits[3]);
    }
}

extern "C" void kernel_launch(void* const* d_in, const int* in_sizes, int n_in,
                              void* d_out, int out_size, void* d_ws, size_t ws_size,
                              hipStream_t stream)
{
    const float* y = (const float*)d_in[0];   // (B, T+MEM) float32
    const float* h = (const float*)d_in[1];   // (MEM,)    float32
    float* out = (float*)d_out;               // (B, T)    float32
    unsigned* dec = (unsigned*)d_ws;          // B * 4097 u32 = ~4.2 MB scratch

    va_forward  <<<dim3(BATCH / 32), dim3(32), 0, stream>>>(y, h, dec);
    va_traceback<<<dim3(BATCH / 32), dim3(32), 0, stream>>>(dec, out);
}